// RGCN_28003186769944
// MI455X (gfx1250) — compile-verified
//
#include <hip/hip_runtime.h>

typedef __attribute__((ext_vector_type(2))) float v2f;
typedef __attribute__((ext_vector_type(8))) float v8f;

// ---------------------------------------------------------------------------
// Edge-weight min/max (monotonic uint key so integer atomics give float order)
// ---------------------------------------------------------------------------
__device__ __forceinline__ unsigned fkey(float f) {
    unsigned u = __float_as_uint(f);
    return (u & 0x80000000u) ? ~u : (u | 0x80000000u);
}
__device__ __forceinline__ float funkey(unsigned k) {
    unsigned u = (k & 0x80000000u) ? (k & 0x7FFFFFFFu) : ~k;
    return __uint_as_float(u);
}

__global__ void mm_init(unsigned* mm) {
    mm[0] = 0xFFFFFFFFu;  // running min key
    mm[1] = 0x00000000u;  // running max key
}

__global__ __launch_bounds__(256) void mm_reduce(const float* __restrict__ ew,
                                                 unsigned* __restrict__ mm, int n) {
    unsigned lo = 0xFFFFFFFFu, hi = 0u;
    for (int i = blockIdx.x * blockDim.x + threadIdx.x; i < n;
         i += gridDim.x * blockDim.x) {
        unsigned k = fkey(ew[i]);
        lo = min(lo, k);
        hi = max(hi, k);
    }
    #pragma unroll
    for (int off = 16; off > 0; off >>= 1) {
        lo = min(lo, (unsigned)__shfl_xor((int)lo, off));
        hi = max(hi, (unsigned)__shfl_xor((int)hi, off));
    }
    if ((threadIdx.x & 31) == 0) {
        atomicMin(&mm[0], lo);
        atomicMax(&mm[1], hi);
    }
}

// ---------------------------------------------------------------------------
// Per-relation weighted aggregation: agg[r][dst] += w_e * x[src]
// One wave per edge; lane handles one float4 (128 feats = 32 lanes * 4).
// ---------------------------------------------------------------------------
__global__ __launch_bounds__(128) void rgcn_agg(const float* __restrict__ x,
                                                const int* __restrict__ src,
                                                const int* __restrict__ dst,
                                                const int* __restrict__ etype,
                                                const float* __restrict__ ew,
                                                const unsigned* __restrict__ mm,
                                                float* __restrict__ agg,
                                                int nedges, int nnodes) {
    const int wave = threadIdx.x >> 5;
    const int lane = threadIdx.x & 31;
    const int e = blockIdx.x * 4 + wave;
    if (e >= nedges) return;

    const float mn = funkey(mm[0]);
    const float mx = funkey(mm[1]);
    const float w = (ew[e] - mn) / (mx - mn + 1e-8f);

    const int s = src[e];
    const int d = dst[e];
    const int r = etype[e];

    const float4 v = ((const float4*)(x + (size_t)s * 128))[lane];
    float* p = agg + ((size_t)r * nnodes + (size_t)d) * 128 + lane * 4;
    unsafeAtomicAdd(p + 0, v.x * w);
    unsafeAtomicAdd(p + 1, v.y * w);
    unsafeAtomicAdd(p + 2, v.z * w);
    unsafeAtomicAdd(p + 3, v.w * w);
}

// ---------------------------------------------------------------------------
// Fused GEMM: out[n, :] = sum_{s=0..3} agg[s][n] @ W[s]  +  xin[n] @ root + b
// Block: 256 thr = 8 waves. Block tile 128 rows x 128 cols; wave tile 16x128.
// K per source (5 x 128) in two 64-wide chunks. The B tile is staged into a
// WMMA-fragment-ready LDS swizzle via GLOBAL_LOAD_ASYNC_TO_LDS_B32 (ASYNCcnt
// background DMA, no VGPR round-trip); fragments then come back with
// conflict-free ds_load_b64.
// ---------------------------------------------------------------------------
__global__ __launch_bounds__(256) void rgcn_gemm(const float* __restrict__ agg,
                                                 const float* __restrict__ xin,
                                                 const float* __restrict__ Wrel,
                                                 const float* __restrict__ root,
                                                 const float* __restrict__ bias,
                                                 float* __restrict__ out,
                                                 int nrows, int relu) {
    // bfrag[g][t][lane][slot]: g = K-group of 4 (16 per 64-chunk),
    // t = 16-col tile (8), lane = 32, slot = 2 -> 8192 floats = 32 KB LDS.
    __shared__ float bfrag[16 * 8 * 32 * 2];

    const int wave = threadIdx.x >> 5;
    const int lane = threadIdx.x & 31;
    const int row0 = blockIdx.x * 128 + wave * 16;

    v8f acc[8];
    #pragma unroll
    for (int t = 0; t < 8; ++t) acc[t] = {};

    // A-fragment row for this lane (clamped for ragged tail; stores predicated)
    const int arow = min(row0 + (lane & 15), nrows - 1);
    const int kofs = (lane >> 4) << 1;  // lanes 16-31 supply K+2, K+3

    // Per-thread staging geometry (invariant across chunks): thread handles
    // column n = tid & 127 for rows k = (tid>>7) + 2*j, j = 0..31.
    const int sn = threadIdx.x & 127;
    const int st = sn >> 4;
    const int sl = sn & 15;
    const int skbase = threadIdx.x >> 7;

    for (int s = 0; s < 5; ++s) {
        const float* B = (s < 4) ? (Wrel + (size_t)s * 128 * 128) : root;
        const float* A = (s < 4) ? (agg + (size_t)s * nrows * 128) : xin;
        const float* Ar = A + (size_t)arow * 128 + kofs;

        for (int kc = 0; kc < 2; ++kc) {  // two K=64 chunks per source
            __syncthreads();  // previous chunk's fragment reads done

            // ---- async DMA: B[kc*64 .. +64)[0..128) -> swizzled LDS ----
            const float* Bc = B + (size_t)kc * 64 * 128;
            #pragma unroll
            for (int j = 0; j < 32; ++j) {
                const int k = skbase + 2 * j;
                const int g = k >> 2;
                const int kk = k & 3;
                // LDS byte offset = low 32 bits of the generic shared address
                const unsigned ldsoff = (unsigned)(uintptr_t)&bfrag[
                    (((g * 8 + st) * 32) + ((kk >> 1) << 4) + sl) * 2 + (kk & 1)];
                const unsigned goff = (unsigned)(((k << 7) + sn) * 4);
                asm volatile(
                    "global_load_async_to_lds_b32 %0, %1, %2"
                    :: "v"(ldsoff), "v"(goff), "s"(Bc)
                    : "memory");
            }
            asm volatile("s_wait_asynccnt 0x0" ::: "memory");
            __syncthreads();

            // ---- compute: preload fragments, then back-to-back WMMAs ----
            const float* Ak = Ar + kc * 64;
            v2f afr[16];
            #pragma unroll
            for (int g = 0; g < 16; ++g)
                afr[g] = *(const v2f*)(Ak + g * 4);

            #pragma unroll
            for (int g = 0; g < 16; ++g) {
                v2f bfr[8];
                #pragma unroll
                for (int t = 0; t < 8; ++t)
                    bfr[t] = *(const v2f*)(&bfrag[((g * 8 + t) * 32 + lane) * 2]);
                #pragma unroll
                for (int t = 0; t < 8; ++t)
                    acc[t] = __builtin_amdgcn_wmma_f32_16x16x4_f32(
                        false, afr[g], false, bfr[t], (short)0, acc[t],
                        false, false);
            }
        }
    }

    // Epilogue: bias (+ ReLU), predicated store.
    // C/D layout: VGPR v -> lanes 0-15: M=v, N=lane; lanes 16-31: M=v+8, N=lane-16
    const int ncol = lane & 15;
    const int rbase = row0 + ((lane >> 4) << 3);
    #pragma unroll
    for (int t = 0; t < 8; ++t) {
        const float bv = bias[t * 16 + ncol];
        #pragma unroll
        for (int v = 0; v < 8; ++v) {
            const int r = rbase + v;
            if (r < nrows) {
                float val = acc[t][v] + bv;
                if (relu) val = fmaxf(val, 0.0f);
                out[(size_t)r * 128 + t * 16 + ncol] = val;
            }
        }
    }
}

// ---------------------------------------------------------------------------
// Host launcher
// Inputs: x, edge_index[2,E], edge_type, edge_weight, W1, root1, b1, W2, root2, b2
// ---------------------------------------------------------------------------
extern "C" void kernel_launch(void* const* d_in, const int* in_sizes, int n_in,
                              void* d_out, int out_size, void* d_ws, size_t ws_size,
                              hipStream_t stream) {
    const float* x     = (const float*)d_in[0];
    const int*   eidx  = (const int*)d_in[1];
    const int*   etype = (const int*)d_in[2];
    const float* ew    = (const float*)d_in[3];
    const float* W1    = (const float*)d_in[4];
    const float* root1 = (const float*)d_in[5];
    const float* b1    = (const float*)d_in[6];
    const float* W2    = (const float*)d_in[7];
    const float* root2 = (const float*)d_in[8];
    const float* b2    = (const float*)d_in[9];

    const int nnodes = in_sizes[0] / 128;
    const int nedges = in_sizes[3];
    const int* src = eidx;
    const int* dst = eidx + nedges;

    // Workspace layout: [minmax: 256B][h: N*128 f32][agg: 4*N*128 f32]
    char* ws = (char*)d_ws;
    unsigned* mm = (unsigned*)ws;
    float* h   = (float*)(ws + 256);
    float* agg = (float*)(ws + 256 + (size_t)nnodes * 128 * sizeof(float));
    const size_t aggbytes = (size_t)4 * nnodes * 128 * sizeof(float);

    mm_init<<<1, 1, 0, stream>>>(mm);
    mm_reduce<<<256, 256, 0, stream>>>(ew, mm, nedges);

    const int ablocks = (nedges + 3) / 4;
    const int gblocks = (nnodes + 127) / 128;

    // Layer 1: aggregate x, GEMM (+ReLU) -> h
    hipMemsetAsync(agg, 0, aggbytes, stream);
    rgcn_agg<<<ablocks, 128, 0, stream>>>(x, src, dst, etype, ew, mm, agg,
                                          nedges, nnodes);
    rgcn_gemm<<<gblocks, 256, 0, stream>>>(agg, x, W1, root1, b1, h, nnodes, 1);

    // Layer 2: aggregate h, GEMM -> out
    hipMemsetAsync(agg, 0, aggbytes, stream);
    rgcn_agg<<<ablocks, 128, 0, stream>>>(h, src, dst, etype, ew, mm, agg,
                                          nedges, nnodes);
    rgcn_gemm<<<gblocks, 256, 0, stream>>>(agg, h, W2, root2, b2, (float*)d_out,
                                           nnodes, 0);
}